// Top2Gate_11940009083381
// MI455X (gfx1250) — compile-verified
//
#include <hip/hip_runtime.h>
#include <hip/hip_bf16.h>
#include <math.h>

typedef __attribute__((ext_vector_type(2))) float v2f;
typedef __attribute__((ext_vector_type(8))) float v8f;

#define S_TOK 8192
#define DDIM  2048
#define NEXP  64
#define CAP   256   // 2 * ceil(8192/64)

// ------------------------------------------------------------------
// Kernel 1: bulk zero-fill of d_out (1.07 GB) with B128 stores.
// ------------------------------------------------------------------
__global__ void fill_kernel(float* __restrict__ out, long long n4, long long ntotal) {
    long long i      = (long long)blockIdx.x * blockDim.x + threadIdx.x;
    long long stride = (long long)gridDim.x * blockDim.x;
    float4 z = make_float4(0.f, 0.f, 0.f, 0.f);
    float4* p = (float4*)out;
    for (; i < n4; i += stride) p[i] = z;
    if (blockIdx.x == 0 && threadIdx.x == 0) {
        for (long long j = n4 * 4; j < ntotal; ++j) out[j] = 0.f;
    }
}

__global__ void init_kernel(float* __restrict__ gsum) {
    if (threadIdx.x < NEXP) gsum[threadIdx.x] = 0.f;
}

// ------------------------------------------------------------------
// Kernel 2: logits[s,e] = sum_d x[s,d] * wg[e,d]  via WMMA f32 16x16x4
// One wave per 16-token row tile; 4 accumulator tiles cover E=64.
// A frag (16x4 f32): lane m=lane&15; lanes<16 hold K=k,k+1; lanes>=16 K=k+2,k+3
// B frag (4x16 f32): column n=lane&15, same K split.
// C/D (16x16 f32): lanes 0-15 -> N=lane, M=vgpr i; lanes 16-31 -> M=i+8.
// ------------------------------------------------------------------
__global__ __launch_bounds__(32) void logits_kernel(const float* __restrict__ x,
                                                    const float* __restrict__ wg,
                                                    float* __restrict__ logits) {
    const int lane  = threadIdx.x;
    const int row0  = blockIdx.x * 16;
    const int m     = lane & 15;
    const int khalf = (lane >> 4) * 2;   // 0 or 2

    v8f acc0 = (v8f)0.f, acc1 = (v8f)0.f, acc2 = (v8f)0.f, acc3 = (v8f)0.f;

    const float* xrow = x  + (size_t)(row0 + m) * DDIM;
    const float* w0   = wg + (size_t)( 0 + m) * DDIM;
    const float* w1   = wg + (size_t)(16 + m) * DDIM;
    const float* w2   = wg + (size_t)(32 + m) * DDIM;
    const float* w3   = wg + (size_t)(48 + m) * DDIM;

    for (int k = 0; k < DDIM; k += 4) {
        v2f a, b0, b1, b2, b3;
        a.x  = xrow[k + khalf];     a.y  = xrow[k + khalf + 1];
        b0.x = w0[k + khalf];       b0.y = w0[k + khalf + 1];
        b1.x = w1[k + khalf];       b1.y = w1[k + khalf + 1];
        b2.x = w2[k + khalf];       b2.y = w2[k + khalf + 1];
        b3.x = w3[k + khalf];       b3.y = w3[k + khalf + 1];
        acc0 = __builtin_amdgcn_wmma_f32_16x16x4_f32(false, a, false, b0, (short)0, acc0, false, false);
        acc1 = __builtin_amdgcn_wmma_f32_16x16x4_f32(false, a, false, b1, (short)0, acc1, false, false);
        acc2 = __builtin_amdgcn_wmma_f32_16x16x4_f32(false, a, false, b2, (short)0, acc2, false, false);
        acc3 = __builtin_amdgcn_wmma_f32_16x16x4_f32(false, a, false, b3, (short)0, acc3, false, false);
    }

    const int n = m;
    const int mbase = (lane >> 4) * 8;
#pragma unroll
    for (int i = 0; i < 8; ++i) {
        size_t r = (size_t)(row0 + mbase + i) * NEXP;
        logits[r +  0 + n] = acc0[i];
        logits[r + 16 + n] = acc1[i];
        logits[r + 32 + n] = acc2[i];
        logits[r + 48 + n] = acc3[i];
    }
}

// ------------------------------------------------------------------
// Kernel 3: per-token softmax + top1/top2 (first-index tie-break) +
// per-expert gate-sum accumulation (for l_aux's "me").
// One wave per token (lane handles experts lane and lane+32); 8 tokens/block.
// ------------------------------------------------------------------
__global__ __launch_bounds__(256) void gate_kernel(const float* __restrict__ logits,
                                                   int*   __restrict__ idx1,
                                                   int*   __restrict__ idx2,
                                                   float* __restrict__ g1raw,
                                                   float* __restrict__ g2raw,
                                                   float* __restrict__ gsum) {
    __shared__ float lsum[NEXP];
    const int lane = threadIdx.x;          // 0..31
    const int w    = threadIdx.y;          // 0..7
    const int tid  = w * 32 + lane;
    if (tid < NEXP) lsum[tid] = 0.f;
    __syncthreads();

    const int s = blockIdx.x * 8 + w;
    const float l0 = logits[(size_t)s * NEXP + lane];
    const float l1 = logits[(size_t)s * NEXP + 32 + lane];

    // row max
    float mx = fmaxf(l0, l1);
#pragma unroll
    for (int off = 16; off > 0; off >>= 1) mx = fmaxf(mx, __shfl_xor(mx, off, 32));

    const float e0 = expf(l0 - mx);
    const float e1 = expf(l1 - mx);
    float z = e0 + e1;
#pragma unroll
    for (int off = 16; off > 0; off >>= 1) z += __shfl_xor(z, off, 32);

    const float gg0 = e0 / z;
    const float gg1 = e1 / z;

    // top-1 argmax over logits (== argmax of gates), first index wins ties
    float v = l0; int ix = lane;
    if (l1 > v) { v = l1; ix = lane + 32; }
#pragma unroll
    for (int off = 16; off > 0; off >>= 1) {
        float ov = __shfl_xor(v, off, 32);
        int   oi = __shfl_xor(ix, off, 32);
        if (ov > v || (ov == v && oi < ix)) { v = ov; ix = oi; }
    }
    const int top1 = ix;
    const float lv1 = v;

    // top-2: mask out top1
    float a0 = (lane        == top1) ? -INFINITY : l0;
    float a1 = ((lane + 32) == top1) ? -INFINITY : l1;
    float v2 = a0; int ix2 = lane;
    if (a1 > v2) { v2 = a1; ix2 = lane + 32; }
#pragma unroll
    for (int off = 16; off > 0; off >>= 1) {
        float ov = __shfl_xor(v2, off, 32);
        int   oi = __shfl_xor(ix2, off, 32);
        if (ov > v2 || (ov == v2 && oi < ix2)) { v2 = ov; ix2 = oi; }
    }

    if (lane == 0) {
        idx1[s]  = top1;
        idx2[s]  = ix2;
        g1raw[s] = expf(lv1 - mx) / z;   // gates[s, top1]
        g2raw[s] = expf(v2  - mx) / z;   // gates[s, top2]
    }

    atomicAdd(&lsum[lane],      gg0);
    atomicAdd(&lsum[lane + 32], gg1);
    __syncthreads();
    if (tid < NEXP) atomicAdd(&gsum[tid], lsum[tid]);
}

// ------------------------------------------------------------------
// Kernel 4: sequential per-expert scan (token order) + l_aux.
// Thread e owns expert e. rank1[s] = #tokens t<s with idx1==idx1[s];
// loc2[s] = count1_total[idx2[s]] + #tokens t<s with idx2==idx2[s].
// ------------------------------------------------------------------
__global__ __launch_bounds__(64) void scan_kernel(const int* __restrict__ idx1,
                                                  const int* __restrict__ idx2,
                                                  const float* __restrict__ gsum,
                                                  int* __restrict__ rank1,
                                                  int* __restrict__ loc2,
                                                  float* __restrict__ out) {
    __shared__ float red[NEXP];
    const int e = threadIdx.x;

    int cnt = 0;
    for (int s = 0; s < S_TOK; ++s) {
        if (idx1[s] == e) { rank1[s] = cnt; ++cnt; }
    }
    const int base = cnt;   // sum(mask1, axis=0)[e] (pre-capacity, per reference)
    int c2 = 0;
    for (int s = 0; s < S_TOK; ++s) {
        if (idx2[s] == e) { loc2[s] = base + c2; ++c2; }
    }

    // l_aux = mean_e(me*ce) * E*E = E * sum_e me*ce
    const float me = gsum[e] / (float)S_TOK;
    const float ce = (float)cnt / (float)S_TOK;
    red[e] = me * ce;
    __syncthreads();
#pragma unroll
    for (int off = 32; off > 0; off >>= 1) {
        if (e < off) red[e] += red[e + off];
        __syncthreads();
    }
    if (e == 0) out[0] = red[0] * (float)NEXP;
}

// ------------------------------------------------------------------
// Kernel 5: capacity drop + renormalize + scatter the <=2 nonzeros/token.
// ------------------------------------------------------------------
__global__ __launch_bounds__(256) void scatter_kernel(const int* __restrict__ idx1,
                                                      const int* __restrict__ idx2,
                                                      const float* __restrict__ g1raw,
                                                      const float* __restrict__ g2raw,
                                                      const int* __restrict__ rank1,
                                                      const int* __restrict__ loc2,
                                                      float* __restrict__ out) {
    const int s = blockIdx.x * blockDim.x + threadIdx.x;
    if (s >= S_TOK) return;

    const int   e1 = idx1[s], e2 = idx2[s];
    const int   r1 = rank1[s], r2 = loc2[s];
    const bool  m1 = (r1 < CAP), m2 = (r2 < CAP);
    const float ga = m1 ? g1raw[s] : 0.f;
    const float gb = m2 ? g2raw[s] : 0.f;
    const float denom = fmaxf(ga + gb, 1.1920929e-7f);   // FLT_EPSILON clip
    const float w1 = ga / denom;
    const float w2 = gb / denom;

    float* cw = out + 1;
    float* dm = out + 1 + (size_t)S_TOK * NEXP * CAP;

    if (m1 && w1 != 0.f) {
        const size_t o = ((size_t)s * NEXP + e1) * CAP + r1;
        cw[o] = w1; dm[o] = 1.f;
    }
    if (m2 && w2 != 0.f) {
        const size_t o = ((size_t)s * NEXP + e2) * CAP + r2;
        cw[o] = w2; dm[o] = 1.f;
    }
}

// ------------------------------------------------------------------
extern "C" void kernel_launch(void* const* d_in, const int* in_sizes, int n_in,
                              void* d_out, int out_size, void* d_ws, size_t ws_size,
                              hipStream_t stream) {
    const float* x  = (const float*)d_in[0];   // [S, D]
    const float* wg = (const float*)d_in[1];   // [E, D]
    float* out = (float*)d_out;

    // workspace layout
    char* ws = (char*)d_ws;
    float* logits = (float*)ws;  ws += sizeof(float) * (size_t)S_TOK * NEXP;
    int*   idx1   = (int*)ws;    ws += sizeof(int) * S_TOK;
    int*   idx2   = (int*)ws;    ws += sizeof(int) * S_TOK;
    float* g1raw  = (float*)ws;  ws += sizeof(float) * S_TOK;
    float* g2raw  = (float*)ws;  ws += sizeof(float) * S_TOK;
    int*   rank1  = (int*)ws;    ws += sizeof(int) * S_TOK;
    int*   loc2   = (int*)ws;    ws += sizeof(int) * S_TOK;
    float* gsum   = (float*)ws;  ws += sizeof(float) * NEXP;

    const long long ntotal = (long long)out_size;          // 1 + 2*S*E*C
    const long long n4     = ntotal / 4;

    // 1) dominant cost: zero 1.07 GB of output with B128 stores (~46 us floor)
    fill_kernel<<<4096, 256, 0, stream>>>(out, n4, ntotal);
    // 2) zero per-expert accumulators
    init_kernel<<<1, 64, 0, stream>>>(gsum);
    // 3) WMMA GEMM: 512 row tiles, one wave each
    logits_kernel<<<S_TOK / 16, 32, 0, stream>>>(x, wg, logits);
    // 4) softmax + top2 per token
    gate_kernel<<<S_TOK / 8, dim3(32, 8), 0, stream>>>(logits, idx1, idx2, g1raw, g2raw, gsum);
    // 5) sequential token-order scan per expert + l_aux -> out[0]
    scan_kernel<<<1, 64, 0, stream>>>(idx1, idx2, gsum, rank1, loc2, out);
    // 6) scatter surviving assignments
    scatter_kernel<<<(S_TOK + 255) / 256, 256, 0, stream>>>(idx1, idx2, g1raw, g2raw,
                                                            rank1, loc2, out);
}